// IntraNet_22771916604016
// MI455X (gfx1250) — compile-verified
//
#include <hip/hip_runtime.h>
#include <hip/hip_bf16.h>

// Problem constants
#define BB 16
#define NN 256
#define FA 64
#define FE 16

typedef __attribute__((ext_vector_type(2))) float v2f;
typedef __attribute__((ext_vector_type(8))) float v8f;

__device__ __forceinline__ v8f wmma_f32(v2f a, v2f b, v8f c) {
  return __builtin_amdgcn_wmma_f32_16x16x4_f32(false, a, false, b, (short)0, c,
                                               false, false);
}

__device__ __forceinline__ float sigm(float v) {
  return 1.f / (1.f + expf(-v));
}

// 4 independent accumulators per k-step -> compiler can clause the 4 B-loads
// and interleave the 4 WMMAs (confirmed pipelining pattern from k_madj).
template <int LDW>
__device__ __forceinline__ void gemm4(const v2f (&afrag)[16],
                                      const float* __restrict__ wbase,
                                      v8f& a0, v8f& a1, v8f& a2, v8f& a3) {
#pragma unroll
  for (int ks = 0; ks < 16; ++ks) {
    v2f b0 = *(const v2f*)(wbase + 4 * ks + (size_t)0  * LDW);
    v2f b1 = *(const v2f*)(wbase + 4 * ks + (size_t)16 * LDW);
    v2f b2 = *(const v2f*)(wbase + 4 * ks + (size_t)32 * LDW);
    v2f b3 = *(const v2f*)(wbase + 4 * ks + (size_t)48 * LDW);
    a0 = wmma_f32(afrag[ks], b0, a0);
    a1 = wmma_f32(afrag[ks], b1, a1);
    a2 = wmma_f32(afrag[ks], b2, a2);
    a3 = wmma_f32(afrag[ks], b3, a3);
  }
}

__device__ __forceinline__ void store4(float* __restrict__ C, int ldc, int row0,
                                       int n0, int lm, int lh,
                                       const float* __restrict__ bias,
                                       const v8f& a0, const v8f& a1,
                                       const v8f& a2, const v8f& a3) {
  float b0 = 0.f, b1 = 0.f, b2 = 0.f, b3 = 0.f;
  if (bias) {
    b0 = bias[n0 + lm];      b1 = bias[n0 + 16 + lm];
    b2 = bias[n0 + 32 + lm]; b3 = bias[n0 + 48 + lm];
  }
#pragma unroll
  for (int v = 0; v < 8; ++v) {
    float* cp = C + (size_t)(row0 + v + 8 * lh) * ldc + n0 + lm;
    cp[0]  = a0[v] + b0;
    cp[16] = a1[v] + b1;
    cp[32] = a2[v] + b2;
    cp[48] = a3[v] + b3;
  }
}

// ---------------------------------------------------------------------------
// Kernel 1: Eagg[b,i,e] = sum_j adj[b,i,j]*edge[b,i,j,e];  deg[b,i] = sum_j adj
// One 256-thread block per (b,i). Each edge row (j) is one 64B cacheline;
// predicating the load on adj!=0 avoids fetching ~85% of the 64 MiB edge tensor.
// ---------------------------------------------------------------------------
__global__ void k_edge_agg(const float* __restrict__ edge,
                           const float* __restrict__ adj,
                           float* __restrict__ Eagg, float* __restrict__ deg) {
  const int bi = blockIdx.x;        // 0..B*N-1
  const int t  = threadIdx.x;       // 0..255
  const int q  = t & 3;             // which float4 chunk of the 16 edge feats
  const int jg = t >> 2;            // 0..63 j-group

  const float* adjrow = adj + (size_t)bi * NN;
  const float* erow   = edge + (size_t)bi * NN * FE;

  float4 acc = make_float4(0.f, 0.f, 0.f, 0.f);
  float dacc = 0.f;
#pragma unroll
  for (int it = 0; it < 4; ++it) {
    const int j = jg + it * 64;
    const float a = adjrow[j];
    if (q == 0) dacc += a;
    if (a != 0.f) {
      const float4 ev = *(const float4*)(erow + j * FE + q * 4);
      acc.x += a * ev.x; acc.y += a * ev.y;
      acc.z += a * ev.z; acc.w += a * ev.w;
    }
  }

  __shared__ float4 sacc[256];
  __shared__ float  sdeg[256];
  sacc[t] = acc;
  sdeg[t] = dacc;
  __syncthreads();
#pragma unroll
  for (int s = 128; s >= 4; s >>= 1) {
    if (t < s) {
      float4 o = sacc[t + s];
      sacc[t].x += o.x; sacc[t].y += o.y; sacc[t].z += o.z; sacc[t].w += o.w;
      sdeg[t] += sdeg[t + s];
    }
    __syncthreads();
  }
  if (t == 0) deg[bi] = sdeg[0];
  if (t < 4) *(float4*)(Eagg + (size_t)bi * FE + t * 4) = sacc[t];
}

// ---------------------------------------------------------------------------
// Kernel 2: all x-side GEMMs in one pass (A-fragments of x loaded once):
//   Aj  = x @ Wm[:, 0:64]^T
//   Aib = x @ Wm[:, 64:128]^T + bm
//   gh  = x @ W_hh^T + b_hh
// One wave per 16-row tile; N-tiles processed 4-at-a-time for load/WMMA ILP.
// WMMA f32 16x16x4 fragment layouts:
//   A: a[v]  = A[row0 + lane%16][k0 + v + 2*(lane/16)]   -> one float2 load
//   B: b[v]  = W [n0  + lane%16][k0 + v + 2*(lane/16)]   (B = W^T)
//   D: D[v]  -> row = v + 8*(lane/16), col = lane%16
// ---------------------------------------------------------------------------
__global__ void k_xw(const float* __restrict__ x,
                     const float* __restrict__ Wm, const float* __restrict__ bm,
                     const float* __restrict__ W_hh,
                     const float* __restrict__ b_hh,
                     float* __restrict__ Aj, float* __restrict__ Aib,
                     float* __restrict__ gh) {
  const int wave = threadIdx.x >> 5;
  const int lane = threadIdx.x & 31;
  const int row0 = (blockIdx.x * 4 + wave) * 16;
  const int lm   = lane & 15;
  const int lh   = lane >> 4;

  v2f afrag[16];
  const float* arow = x + (size_t)(row0 + lm) * FA + 2 * lh;
#pragma unroll
  for (int ks = 0; ks < 16; ++ks) afrag[ks] = *(const v2f*)(arow + 4 * ks);

  // --- Aj = x @ Wm1^T (one group of 4 n-tiles) ---
  {
    v8f a0 = {}, a1 = {}, a2 = {}, a3 = {};
    gemm4<144>(afrag, Wm + (size_t)lm * 144 + 0 + 2 * lh, a0, a1, a2, a3);
    store4(Aj, FA, row0, 0, lm, lh, nullptr, a0, a1, a2, a3);
  }
  // --- Aib = x @ Wm2^T + bm ---
  {
    v8f a0 = {}, a1 = {}, a2 = {}, a3 = {};
    gemm4<144>(afrag, Wm + (size_t)lm * 144 + 64 + 2 * lh, a0, a1, a2, a3);
    store4(Aib, FA, row0, 0, lm, lh, bm, a0, a1, a2, a3);
  }
  // --- gh = x @ W_hh^T + b_hh (3 groups of 4 n-tiles, N = 192) ---
#pragma unroll
  for (int grp = 0; grp < 3; ++grp) {
    const int n0 = grp * 64;
    v8f a0 = {}, a1 = {}, a2 = {}, a3 = {};
    gemm4<FA>(afrag, W_hh + (size_t)(n0 + lm) * FA + 2 * lh, a0, a1, a2, a3);
    store4(gh, 192, row0, n0, lm, lh, b_hh, a0, a1, a2, a3);
  }
}

// ---------------------------------------------------------------------------
// Kernel 3: m = Eagg @ Wm[:,128:144]^T  +  adj[b] @ Aj[b]  +  deg * Aib
// Accumulators are seeded by the K=16 edge-feature WMMA, then accumulate the
// K=256 neighbor GEMM; the deg*Aib term is added in the epilogue.
// One wave per (batch, 16-row tile): 256 waves -> 64 blocks.
// ---------------------------------------------------------------------------
__global__ void k_madj_fused(const float* __restrict__ adj,
                             const float* __restrict__ Aj,
                             const float* __restrict__ Eagg,
                             const float* __restrict__ Wm,
                             const float* __restrict__ Aib,
                             const float* __restrict__ deg,
                             float* __restrict__ m) {
  const int wave = threadIdx.x >> 5;
  const int lane = threadIdx.x & 31;
  const int gt   = blockIdx.x * 4 + wave;   // 0..255
  const int b    = gt >> 4;
  const int mt   = gt & 15;
  const int lm   = lane & 15;
  const int lh   = lane >> 4;

  const int row0  = mt * 16;                 // row within batch
  const int grow0 = b * NN + row0;           // global row

  v8f acc0 = {}, acc1 = {}, acc2 = {}, acc3 = {};

  // Seed with EW = Eagg @ Wm3^T (K = 16)
  {
    const float* erow = Eagg + (size_t)(grow0 + lm) * FE + 2 * lh;
#pragma unroll
    for (int ks = 0; ks < 4; ++ks) {
      v2f a = *(const v2f*)(erow + 4 * ks);
      const float* w = Wm + (size_t)lm * 144 + 128 + 4 * ks + 2 * lh;
      acc0 = wmma_f32(a, *(const v2f*)(w + (size_t)0  * 144), acc0);
      acc1 = wmma_f32(a, *(const v2f*)(w + (size_t)16 * 144), acc1);
      acc2 = wmma_f32(a, *(const v2f*)(w + (size_t)32 * 144), acc2);
      acc3 = wmma_f32(a, *(const v2f*)(w + (size_t)48 * 144), acc3);
    }
  }

  // Accumulate adj[b] @ Aj[b]  (K = 256)
  const float* Ab = adj + (size_t)b * NN * NN;
  const float* Bb = Aj  + (size_t)b * NN * FA;
  const float* arow = Ab + (size_t)(row0 + lm) * NN + 2 * lh;
  for (int k0 = 0; k0 < NN; k0 += 4) {
    v2f a = *(const v2f*)(arow + k0);
    const float* bp = Bb + (size_t)(k0 + 2 * lh) * FA + lm;   // B[k][n]
    v2f b0, b1, b2, b3;
    b0[0] = bp[0];  b0[1] = bp[FA + 0];
    b1[0] = bp[16]; b1[1] = bp[FA + 16];
    b2[0] = bp[32]; b2[1] = bp[FA + 32];
    b3[0] = bp[48]; b3[1] = bp[FA + 48];
    acc0 = wmma_f32(a, b0, acc0);
    acc1 = wmma_f32(a, b1, acc1);
    acc2 = wmma_f32(a, b2, acc2);
    acc3 = wmma_f32(a, b3, acc3);
  }

  // Epilogue: m = acc + deg * Aib
#pragma unroll
  for (int v = 0; v < 8; ++v) {
    const int g = grow0 + v + 8 * lh;
    const float dv = deg[g];
    const size_t base = (size_t)g * FA + lm;
    m[base + 0 ] = acc0[v] + dv * Aib[base + 0 ];
    m[base + 16] = acc1[v] + dv * Aib[base + 16];
    m[base + 32] = acc2[v] + dv * Aib[base + 32];
    m[base + 48] = acc3[v] + dv * Aib[base + 48];
  }
}

// ---------------------------------------------------------------------------
// Kernel 4: gi = m @ W_ih^T + b_ih fused with GRU gates -> out.
// For each 16-column group t, the r/z/n tiles (cols t, 64+t, 128+t) are
// computed together (3 independent accumulators per k-step) so gate math
// runs elementwise on the D-fragments.
// ---------------------------------------------------------------------------
__global__ void k_gi_gates(const float* __restrict__ m,
                           const float* __restrict__ W_ih,
                           const float* __restrict__ b_ih,
                           const float* __restrict__ gh,
                           const float* __restrict__ x,
                           float* __restrict__ out) {
  const int wave = threadIdx.x >> 5;
  const int lane = threadIdx.x & 31;
  const int row0 = (blockIdx.x * 4 + wave) * 16;
  const int lm   = lane & 15;
  const int lh   = lane >> 4;

  v2f afrag[16];
  const float* arow = m + (size_t)(row0 + lm) * FA + 2 * lh;
#pragma unroll
  for (int ks = 0; ks < 16; ++ks) afrag[ks] = *(const v2f*)(arow + 4 * ks);

#pragma unroll
  for (int t = 0; t < 4; ++t) {
    const int n0 = 16 * t;
    v8f ar = {}, az = {}, an = {};
    const float* wr = W_ih + (size_t)(n0 + lm) * FA + 2 * lh;
#pragma unroll
    for (int ks = 0; ks < 16; ++ks) {
      v2f a  = afrag[ks];
      v2f b0 = *(const v2f*)(wr + 4 * ks + (size_t)0   * FA);
      v2f b1 = *(const v2f*)(wr + 4 * ks + (size_t)64  * FA);
      v2f b2 = *(const v2f*)(wr + 4 * ks + (size_t)128 * FA);
      ar = wmma_f32(a, b0, ar);
      az = wmma_f32(a, b1, az);
      an = wmma_f32(a, b2, an);
    }
    const float br = b_ih[n0 + lm];
    const float bz = b_ih[64 + n0 + lm];
    const float bn = b_ih[128 + n0 + lm];
#pragma unroll
    for (int v = 0; v < 8; ++v) {
      const int row = row0 + v + 8 * lh;
      const int col = n0 + lm;
      const float* ghr = gh + (size_t)row * 192 + col;
      const float hr = ghr[0], hz = ghr[64], hn = ghr[128];
      const float r  = sigm(ar[v] + br + hr);
      const float z  = sigm(az[v] + bz + hz);
      const float nc = tanhf(an[v] + bn + r * hn);
      const float xv = x[(size_t)row * FA + col];
      out[(size_t)row * FA + col] = (1.f - z) * nc + z * xv;
    }
  }
}

// ---------------------------------------------------------------------------
extern "C" void kernel_launch(void* const* d_in, const int* in_sizes, int n_in,
                              void* d_out, int out_size, void* d_ws,
                              size_t ws_size, hipStream_t stream) {
  (void)in_sizes; (void)n_in; (void)out_size; (void)ws_size;
  const float* edge = (const float*)d_in[0];  // [16,256,256,16]
  const float* adj  = (const float*)d_in[1];  // [16,256,256]
  const float* x    = (const float*)d_in[2];  // [16,256,64]
  const float* Wm   = (const float*)d_in[3];  // [64,144]
  const float* bm   = (const float*)d_in[4];  // [64]
  const float* W_ih = (const float*)d_in[5];  // [192,64]
  const float* W_hh = (const float*)d_in[6];  // [192,64]
  const float* b_ih = (const float*)d_in[7];  // [192]
  const float* b_hh = (const float*)d_in[8];  // [192]
  float* out = (float*)d_out;                 // [16,256,64]

  const int M = BB * NN;                      // 4096 rows
  float* ws = (float*)d_ws;
  float* Eagg = ws;                 // 4096*16
  float* deg  = Eagg + M * FE;      // 4096
  float* Aj   = deg + M;            // 4096*64
  float* Aib  = Aj + M * FA;        // 4096*64
  float* gh   = Aib + M * FA;       // 4096*192
  float* mbuf = gh + M * 192;       // 4096*64

  // 1. edge aggregation + degree (bandwidth-bound term)
  k_edge_agg<<<M, 256, 0, stream>>>(edge, adj, Eagg, deg);
  // 2. all x-side GEMMs (Aj, Aib+bm, gh+b_hh)
  k_xw<<<64, 128, 0, stream>>>(x, Wm, bm, W_hh, b_hh, Aj, Aib, gh);
  // 3. m = Eagg@Wm3^T + adj[b]@Aj[b] + deg*Aib
  k_madj_fused<<<64, 128, 0, stream>>>(adj, Aj, Eagg, Wm, Aib, deg, mbuf);
  // 4. gi GEMM + GRU gates -> out
  k_gi_gates<<<64, 128, 0, stream>>>(mbuf, W_ih, b_ih, gh, x, out);
}